// CAWformer_56229711839932
// MI455X (gfx1250) — compile-verified
//
#include <hip/hip_runtime.h>
#include <math.h>

#define Bn 16
#define Tn 512
#define Cn 128
#define DMn 512
#define Ln 3
#define Pn 64
#define KSn 25
#define NSHIFT 8
#define EPSf 1e-5f

typedef __attribute__((ext_vector_type(2))) float v2f;
typedef __attribute__((ext_vector_type(8))) float v8f;

__device__ __forceinline__ float waveSum(float v) {
#pragma unroll
  for (int off = 16; off > 0; off >>= 1) v += __shfl_down(v, off, 32);
  return v;
}
__device__ __forceinline__ float waveMax(float v) {
#pragma unroll
  for (int off = 16; off > 0; off >>= 1) v = fmaxf(v, __shfl_down(v, off, 32));
  return v;
}

// ---------------------------------------------------------------------------
// Depthwise smoothing conv (replicate pad) + residual, writing (B, C, T).
// ---------------------------------------------------------------------------
__global__ void conv_residual_kernel(const float* __restrict__ inp,
                                     const float* __restrict__ w,
                                     const float* __restrict__ bptr,
                                     float* __restrict__ resid) {
  __shared__ float sx[Tn];
  __shared__ float sw[KSn];
  const int bc = blockIdx.x;  // b*Cn + c
  const int b = bc / Cn, c = bc % Cn;
  const int tid = threadIdx.x;  // 256
  for (int t = tid; t < Tn; t += 256) sx[t] = inp[((size_t)b * Tn + t) * Cn + c];
  if (tid < KSn) sw[tid] = w[tid];
  __syncthreads();
  const float bias = bptr[0];
  for (int t = tid; t < Tn; t += 256) {
    float acc = bias;
#pragma unroll
    for (int j = 0; j < KSn; ++j) {
      int tt = t + j - (KSn / 2);
      tt = tt < 0 ? 0 : (tt > Tn - 1 ? Tn - 1 : tt);
      acc += sw[j] * sx[tt];
    }
    resid[(size_t)bc * Tn + t] = sx[t] - acc;
  }
}

// ---------------------------------------------------------------------------
// Wrap-free inner GEMM segment over logical k in [kStart, kEnd).
// Abase is pre-offset so the lane's A element is Abase[k(+1)] directly.
// WNXK=1: B[k,n]=W[n*ldw+k] (weight as NxK, vector b64 loads).
// WNXK=0: B[k,n]=W[k*ldw+n] (row-major activations, scalar loads).
// ---------------------------------------------------------------------------
template <int WNXK>
__device__ __forceinline__ void gemm_seg(
    const float* __restrict__ Abase, const float* __restrict__ W, int ldw,
    int n0, int half, int kStart, int kEnd,
    v8f& acc0, v8f& acc1, v8f& acc2, v8f& acc3) {
#pragma unroll 4
  for (int k0 = kStart; k0 < kEnd; k0 += 4) {
    const int kk = k0 + 2 * half;  // ISA 7.12.2: A 16x4 -> lane-half picks K pair
    const v2f a = *(const v2f*)(Abase + kk);
    v2f b0, b1, b2, b3;
    if (WNXK) {
      const float* W0 = W + (size_t)n0 * ldw + kk;
      b0 = *(const v2f*)(W0);
      b1 = *(const v2f*)(W0 + (size_t)16 * ldw);
      b2 = *(const v2f*)(W0 + (size_t)32 * ldw);
      b3 = *(const v2f*)(W0 + (size_t)48 * ldw);
    } else {
      const float* W0 = W + (size_t)kk * ldw + n0;
      b0.x = W0[0];   b0.y = W0[ldw];
      b1.x = W0[16];  b1.y = W0[16 + ldw];
      b2.x = W0[32];  b2.y = W0[32 + ldw];
      b3.x = W0[48];  b3.y = W0[48 + ldw];
    }
    acc0 = __builtin_amdgcn_wmma_f32_16x16x4_f32(false, a, false, b0, (short)0, acc0, false, false);
    acc1 = __builtin_amdgcn_wmma_f32_16x16x4_f32(false, a, false, b1, (short)0, acc1, false, false);
    acc2 = __builtin_amdgcn_wmma_f32_16x16x4_f32(false, a, false, b2, (short)0, acc2, false, false);
    acc3 = __builtin_amdgcn_wmma_f32_16x16x4_f32(false, a, false, b3, (short)0, acc3, false, false);
  }
}

// ---------------------------------------------------------------------------
// Generic fp32 WMMA GEMM:  O = epilogue( A @ B )  using V_WMMA_F32_16X16X4_F32
//   A[m,k'] with k' = (k + rot) % K, implemented as two wrap-free segments.
// Block = 128 threads (4 waves), tile 64x64; wave holds 4 x (16x16 f32 acc).
// Epilogue: +bias[n], +pos[(m%Cmod)*ldpos+n], +res[m*ldr+n], exact GELU,
//           eval-BN: v*(bnG[m%Cmod]*bnS)+bnB[m%Cmod].
// All problem dims here are multiples of 64/4, so no bounds checks needed.
// ---------------------------------------------------------------------------
template <int WNXK>
__global__ void gemm_wmma_kernel(
    const float* __restrict__ A, int lda, long aBS, int rotBase, int rotStep,
    const float* __restrict__ W, int ldw, long wBS,
    float* __restrict__ O, int ldo, long oBS,
    const float* __restrict__ bias,
    const float* __restrict__ res, int ldr, long rBS,
    const float* __restrict__ pos, int ldpos,
    const float* __restrict__ bnG, const float* __restrict__ bnB, float bnS,
    int doGelu, int Cmod, int K) {
  const int z = blockIdx.z;
  A += (size_t)z * aBS;
  W += (size_t)z * wBS;
  O += (size_t)z * oBS;
  if (res) res += (size_t)z * rBS;
  const int rot = rotBase + rotStep * z;  // in [0, K)

  const int lane = threadIdx.x & 31;
  const int wv = threadIdx.x >> 5;
  const int half = lane >> 4;
  const int l16 = lane & 15;

  const int mRow = blockIdx.y * 64 + wv * 16 + l16;  // row this lane feeds to A-frag
  const int n0 = blockIdx.x * 64 + l16;              // base col for B-frag loads

  v8f acc0 = {}, acc1 = {}, acc2 = {}, acc3 = {};
  const float* Arow = A + (size_t)mRow * lda;

  const int kSplit = K - rot;  // logical k < kSplit reads Arow[k+rot]
  gemm_seg<WNXK>(Arow + rot, W, ldw, n0, half, 0, kSplit, acc0, acc1, acc2, acc3);
  if (rot > 0)
    gemm_seg<WNXK>(Arow + rot - K, W, ldw, n0, half, kSplit, K, acc0, acc1, acc2, acc3);

  // C/D layout (ISA 7.12.2): VGPR r, lanes 0-15 -> M=r, lanes 16-31 -> M=8+r; N=lane%16.
  const int mBase = blockIdx.y * 64 + wv * 16 + half * 8;
  const v8f accs[4] = {acc0, acc1, acc2, acc3};
#pragma unroll
  for (int t = 0; t < 4; ++t) {
    const int n = blockIdx.x * 64 + t * 16 + l16;
    const float bval = bias ? bias[n] : 0.0f;
#pragma unroll
    for (int r = 0; r < 8; ++r) {
      const int m = mBase + r;
      float v = accs[t][r] + bval;
      if (pos) v += pos[(size_t)(m % Cmod) * ldpos + n];
      if (res) v += res[(size_t)m * ldr + n];
      if (doGelu) v = 0.5f * v * (1.0f + erff(v * 0.70710678118654752f));
      if (bnG) {
        const int c = m % Cmod;
        v = v * (bnG[c] * bnS) + (bnB ? bnB[c] : 0.0f);
      }
      O[(size_t)m * ldo + n] = v;
    }
  }
}

// ---------------------------------------------------------------------------
// Row sums of X (rows = B*C, width DM). One wave per row.
// ---------------------------------------------------------------------------
__global__ void rowsum_kernel(const float* __restrict__ X, float* __restrict__ S) {
  const int lane = threadIdx.x & 31, wv = threadIdx.x >> 5;
  const int row = blockIdx.x * 4 + wv;
  const float* x = X + (size_t)row * DMn;
  float s = 0.0f;
  for (int d = lane; d < DMn; d += 32) s += x[d];
  s = waveSum(s);
  if (lane == 0) S[row] = s;
}

// ---------------------------------------------------------------------------
// corr[b,i,j] = s_i*s_j / (DM*sqrt(DM)); softmax over j.  (analytic FFT collapse)
// Block per (b,i), 128 threads (one per j).
// ---------------------------------------------------------------------------
__global__ void corr_softmax_kernel(const float* __restrict__ S, float* __restrict__ Pm) {
  const int bi = blockIdx.x;
  const int b = bi / Cn;
  const int i = bi % Cn;
  const int j = threadIdx.x;
  const float sc = 1.0f / (512.0f * 22.627416997969522f);  // DM^-1.5
  const float e = S[b * Cn + i] * S[b * Cn + j] * sc;
  __shared__ float sm[4];
  const int lane = j & 31, wv = j >> 5;
  float m = waveMax(e);
  if (lane == 0) sm[wv] = m;
  __syncthreads();
  const float mm = fmaxf(fmaxf(sm[0], sm[1]), fmaxf(sm[2], sm[3]));
  const float ex = expf(e - mm);
  __syncthreads();
  float s2 = waveSum(ex);
  if (lane == 0) sm[wv] = s2;
  __syncthreads();
  const float tot = sm[0] + sm[1] + sm[2] + sm[3];
  Pm[(size_t)bi * Cn + j] = ex / tot;
}

// ---------------------------------------------------------------------------
// LayerNorm over DM=512 per row. 256 threads, 2 elems/thread.
// ---------------------------------------------------------------------------
__global__ void layernorm_kernel(const float* __restrict__ In, float* __restrict__ Out,
                                 const float* __restrict__ g, const float* __restrict__ bta) {
  const int row = blockIdx.x;
  const int tid = threadIdx.x;
  const int lane = tid & 31, wv = tid >> 5;
  const float* x = In + (size_t)row * DMn;
  const float x0 = x[tid], x1 = x[tid + 256];
  __shared__ float sm[8], sm2[8];
  float s = waveSum(x0 + x1);
  float q = waveSum(x0 * x0 + x1 * x1);
  if (lane == 0) { sm[wv] = s; sm2[wv] = q; }
  __syncthreads();
  float tot = 0.0f, tq = 0.0f;
#pragma unroll
  for (int i = 0; i < 8; ++i) { tot += sm[i]; tq += sm2[i]; }
  const float mu = tot * (1.0f / DMn);
  const float var = tq * (1.0f / DMn) - mu * mu;
  const float rs = rsqrtf(var + EPSf);
  Out[(size_t)row * DMn + tid]       = (x0 - mu) * rs * g[tid] + bta[tid];
  Out[(size_t)row * DMn + tid + 256] = (x1 - mu) * rs * g[tid + 256] + bta[tid + 256];
}

// ---------------------------------------------------------------------------
// Auto-attn combine: per row (b,c): scores[s] = (q . k_s)/sqrt(DM), softmax
// over the 8 shifts, O = sum_s p_s * v_s.  8 waves/block, wave s owns shift s.
// ---------------------------------------------------------------------------
__global__ void attn_combine_kernel(const float* __restrict__ Q,
                                    const float* __restrict__ Kb,
                                    const float* __restrict__ Vb,
                                    float* __restrict__ O) {
  const int row = blockIdx.x;
  const int tid = threadIdx.x;  // 256 = 8 waves
  const int lane = tid & 31, s = tid >> 5;
  const float* q = Q + (size_t)row * DMn;
  const float* k = Kb + (size_t)row * NSHIFT * DMn + (size_t)s * DMn;
  __shared__ float sc[NSHIFT];
  float acc = 0.0f;
  for (int d = lane; d < DMn; d += 32) acc += q[d] * k[d];
  acc = waveSum(acc);
  if (lane == 0) sc[s] = acc * 0.044194173824159216f;  // DM^-0.5
  __syncthreads();
  float mx = sc[0];
#pragma unroll
  for (int i = 1; i < NSHIFT; ++i) mx = fmaxf(mx, sc[i]);
  float p[NSHIFT];
  float sum = 0.0f;
#pragma unroll
  for (int i = 0; i < NSHIFT; ++i) { p[i] = expf(sc[i] - mx); sum += p[i]; }
  const float inv = 1.0f / sum;
  const float* v = Vb + (size_t)row * NSHIFT * DMn;
  for (int d = tid; d < DMn; d += 256) {
    float o = 0.0f;
#pragma unroll
    for (int i = 0; i < NSHIFT; ++i) o += p[i] * v[(size_t)i * DMn + d];
    O[(size_t)row * DMn + d] = o * inv;
  }
}

// ---------------------------------------------------------------------------
extern "C" void kernel_launch(void* const* d_in, const int* in_sizes, int n_in,
                              void* d_out, int out_size, void* d_ws, size_t ws_size,
                              hipStream_t stream) {
  (void)in_sizes; (void)n_in; (void)out_size; (void)ws_size;
  const float* inp     = (const float*)d_in[0];
  const float* ld_w    = (const float*)d_in[1];
  const float* ld_b    = (const float*)d_in[2];
  const float* emb_W   = (const float*)d_in[3];
  const float* emb_b   = (const float*)d_in[4];
  const float* W_pos   = (const float*)d_in[5];
  const float* vc_bn_g = (const float*)d_in[6];
  const float* vc_bn_b = (const float*)d_in[7];
  const float* vc_ln_g = (const float*)d_in[8];
  const float* vc_ln_b = (const float*)d_in[9];
  const float* vc_W1   = (const float*)d_in[10];
  const float* vc_b1   = (const float*)d_in[11];
  const float* vc_W2   = (const float*)d_in[12];
  const float* vc_b2   = (const float*)d_in[13];
  const float* aa_Wq   = (const float*)d_in[14];
  const float* aa_bq   = (const float*)d_in[15];
  const float* aa_Wk   = (const float*)d_in[16];
  const float* aa_bk   = (const float*)d_in[17];
  const float* aa_Wv   = (const float*)d_in[18];
  const float* aa_bv   = (const float*)d_in[19];
  const float* aa_Wo   = (const float*)d_in[20];
  const float* aa_bo   = (const float*)d_in[21];
  const float* aa_bn_g = (const float*)d_in[22];
  const float* aa_bn_b = (const float*)d_in[23];
  const float* aa_ln_g = (const float*)d_in[24];
  const float* aa_ln_b = (const float*)d_in[25];
  const float* aa_W1   = (const float*)d_in[26];
  const float* aa_b1   = (const float*)d_in[27];
  const float* aa_W2   = (const float*)d_in[28];
  const float* aa_b2   = (const float*)d_in[29];

  const size_t NROW = (size_t)Bn * Cn;  // 2048
  float* ws = (float*)d_ws;
  float* resid = ws; ws += NROW * Tn;
  float* X     = ws; ws += NROW * DMn;
  float* R2    = ws; ws += NROW * DMn;
  float* H     = ws; ws += NROW * 2 * DMn;
  float* Sv    = ws; ws += NROW;
  float* Pm    = ws; ws += (size_t)Bn * Cn * Cn;
  float* Qb    = ws; ws += NROW * DMn;
  float* Ab    = ws; ws += NROW * DMn;
  float* Kb    = ws; ws += NROW * NSHIFT * DMn;
  float* Vb    = ws; ws += NROW * NSHIFT * DMn;

  const float bnS = 1.0f / sqrtf(1.0f + 1e-5f);
  const dim3 blk(128);

  conv_residual_kernel<<<dim3(Bn * Cn), dim3(256), 0, stream>>>(inp, ld_w, ld_b, resid);

  // Embedding: X = resid @ emb_W^T + emb_b + W_pos   (M=2048,K=T,N=DM)
  gemm_wmma_kernel<1><<<dim3(DMn / 64, 2048 / 64, 1), blk, 0, stream>>>(
      resid, Tn, 0, 0, 0,
      emb_W, Tn, 0,
      X, DMn, 0,
      emb_b,
      nullptr, 0, 0,
      W_pos, DMn,
      nullptr, nullptr, 1.0f,
      0, Cn, Tn);

  for (int l = 0; l < Ln; ++l) {
    const float* bnG1 = vc_bn_g + l * Cn;  const float* bnB1 = vc_bn_b + l * Cn;
    const float* lnG1 = vc_ln_g + l * DMn; const float* lnB1 = vc_ln_b + l * DMn;
    const float* W1a = vc_W1 + (size_t)l * 2 * DMn * DMn; const float* b1a = vc_b1 + l * 2 * DMn;
    const float* W2a = vc_W2 + (size_t)l * DMn * 2 * DMn; const float* b2a = vc_b2 + l * DMn;
    const float* Wq = aa_Wq + (size_t)l * DMn * DMn; const float* bq = aa_bq + l * DMn;
    const float* Wk = aa_Wk + (size_t)l * DMn * DMn; const float* bk = aa_bk + l * DMn;
    const float* Wv = aa_Wv + (size_t)l * DMn * DMn; const float* bv = aa_bv + l * DMn;
    const float* Wo = aa_Wo + (size_t)l * DMn * DMn; const float* bo = aa_bo + l * DMn;
    const float* bnG2 = aa_bn_g + l * Cn;  const float* bnB2 = aa_bn_b + l * Cn;
    const float* lnG2 = aa_ln_g + l * DMn; const float* lnB2 = aa_ln_b + l * DMn;
    const float* W1b = aa_W1 + (size_t)l * 2 * DMn * DMn; const float* b1b = aa_b1 + l * 2 * DMn;
    const float* W2b = aa_W2 + (size_t)l * DMn * 2 * DMn; const float* b2b = aa_b2 + l * DMn;

    // ---- VarCor block (FFT collapsed to row-sum rank-1 softmax) ----
    rowsum_kernel<<<dim3(2048 / 4), blk, 0, stream>>>(X, Sv);
    corr_softmax_kernel<<<dim3(Bn * Cn), dim3(Cn), 0, stream>>>(Sv, Pm);
    // R2 = BN( P @ X + X )   batched over B
    gemm_wmma_kernel<0><<<dim3(DMn / 64, Cn / 64, Bn), blk, 0, stream>>>(
        Pm, Cn, (long)Cn * Cn, 0, 0,
        X, DMn, (long)Cn * DMn,
        R2, DMn, (long)Cn * DMn,
        nullptr,
        X, DMn, (long)Cn * DMn,
        nullptr, 0,
        bnG1, bnB1, bnS,
        0, Cn, Cn);
    // H = gelu(R2 @ W1^T + b1)
    gemm_wmma_kernel<1><<<dim3(2 * DMn / 64, 2048 / 64, 1), blk, 0, stream>>>(
        R2, DMn, 0, 0, 0,
        W1a, DMn, 0,
        H, 2 * DMn, 0,
        b1a,
        nullptr, 0, 0, nullptr, 0,
        nullptr, nullptr, 1.0f,
        1, Cn, DMn);
    // X = H @ W2^T + b2 + R2  (pre-LN)
    gemm_wmma_kernel<1><<<dim3(DMn / 64, 2048 / 64, 1), blk, 0, stream>>>(
        H, 2 * DMn, 0, 0, 0,
        W2a, 2 * DMn, 0,
        X, DMn, 0,
        b2a,
        R2, DMn, 0, nullptr, 0,
        nullptr, nullptr, 1.0f,
        0, Cn, 2 * DMn);
    layernorm_kernel<<<dim3(2048), dim3(256), 0, stream>>>(X, X, lnG1, lnB1);

    // ---- Auto-attention block ----
    // Q = X @ Wq^T + bq
    gemm_wmma_kernel<1><<<dim3(DMn / 64, 2048 / 64, 1), blk, 0, stream>>>(
        X, DMn, 0, 0, 0,
        Wq, DMn, 0,
        Qb, DMn, 0,
        bq, nullptr, 0, 0, nullptr, 0, nullptr, nullptr, 1.0f, 0, Cn, DMn);
    // K/V for all 8 circular shifts in one batched launch (rot = 64*z on A cols)
    gemm_wmma_kernel<1><<<dim3(DMn / 64, 2048 / 64, NSHIFT), blk, 0, stream>>>(
        X, DMn, 0, 0, Pn,
        Wk, DMn, 0,
        Kb, NSHIFT * DMn, (long)DMn,
        bk, nullptr, 0, 0, nullptr, 0, nullptr, nullptr, 1.0f, 0, Cn, DMn);
    gemm_wmma_kernel<1><<<dim3(DMn / 64, 2048 / 64, NSHIFT), blk, 0, stream>>>(
        X, DMn, 0, 0, Pn,
        Wv, DMn, 0,
        Vb, NSHIFT * DMn, (long)DMn,
        bv, nullptr, 0, 0, nullptr, 0, nullptr, nullptr, 1.0f, 0, Cn, DMn);
    attn_combine_kernel<<<dim3(2048), dim3(256), 0, stream>>>(Qb, Kb, Vb, Ab);
    // R2 = BN( Ab @ Wo^T + bo + X )
    gemm_wmma_kernel<1><<<dim3(DMn / 64, 2048 / 64, 1), blk, 0, stream>>>(
        Ab, DMn, 0, 0, 0,
        Wo, DMn, 0,
        R2, DMn, 0,
        bo,
        X, DMn, 0, nullptr, 0,
        bnG2, bnB2, bnS,
        0, Cn, DMn);
    // H = gelu(R2 @ W1^T + b1)
    gemm_wmma_kernel<1><<<dim3(2 * DMn / 64, 2048 / 64, 1), blk, 0, stream>>>(
        R2, DMn, 0, 0, 0,
        W1b, DMn, 0,
        H, 2 * DMn, 0,
        b1b, nullptr, 0, 0, nullptr, 0, nullptr, nullptr, 1.0f, 1, Cn, DMn);
    // X = H @ W2^T + b2 + R2
    gemm_wmma_kernel<1><<<dim3(DMn / 64, 2048 / 64, 1), blk, 0, stream>>>(
        H, 2 * DMn, 0, 0, 0,
        W2b, 2 * DMn, 0,
        X, DMn, 0,
        b2b, R2, DMn, 0, nullptr, 0, nullptr, nullptr, 1.0f, 0, Cn, 2 * DMn);
    float* lnOut = (l == Ln - 1) ? (float*)d_out : X;
    layernorm_kernel<<<dim3(2048), dim3(256), 0, stream>>>(X, lnOut, lnG2, lnB2);
  }
}